// AngleTripletGenerator_15607911153873
// MI455X (gfx1250) — compile-verified
//
#include <hip/hip_runtime.h>
#include <hip/hip_bf16.h>
#include <math.h>

// CDNA5 (gfx1250) wave32 WMMA types
typedef __attribute__((ext_vector_type(2))) float v2f;   // 16x4 f32 A/B operand: 2 VGPRs
typedef __attribute__((ext_vector_type(8))) float v8f;   // 16x16 f32 C/D: 8 VGPRs

#define DEG      16
#define CUTOFF2  25.0f   // CUTOFF^2; dist <= 5  <=>  dist^2 <= 25

// One wave32 per center node. The wave computes the 16x16 Gram matrix
// dot[j,k] = R1_j . R1_k with a single V_WMMA_F32_16X16X4_F32 (K padded 3->4),
// then derives angles and pair distances from the Gram identity.
__global__ __launch_bounds__(256)
void triplet_wmma_kernel(const float* __restrict__ pos,        // [N,3]
                         const int*   __restrict__ edge_index, // [2,E]
                         int*   __restrict__ id3_i,
                         int*   __restrict__ id3_j,
                         int*   __restrict__ id3_k,
                         float* __restrict__ dist_out,
                         float* __restrict__ ang_out,
                         unsigned char* __restrict__ mask_out,
                         int num_nodes)
{
    const int n = blockIdx.x * (blockDim.x >> 5) + (threadIdx.x >> 5);
    if (n >= num_nodes) return;            // wave-uniform: EXEC stays all-1s for WMMA

    const int  lane = threadIdx.x & 31;    // wave32
    const int  slot = lane & 15;           // edge slot this lane owns (both halves)
    const bool hihalf = lane >= 16;
    const int  E = num_nodes * DEG;

    // Gather this slot's neighbor displacement R1 = pos[nb] - pos[n]
    const int nbr = edge_index[E + n * DEG + slot];   // row 1 of edge_index
    const float cx = pos[3 * n + 0];
    const float cy = pos[3 * n + 1];
    const float cz = pos[3 * n + 2];
    const float dx = pos[3 * nbr + 0] - cx;
    const float dy = pos[3 * nbr + 1] - cy;
    const float dz = pos[3 * nbr + 2] - cz;
    const float sq  = dx * dx + dy * dy + dz * dz;    // |R1|^2
    const bool  val = (sq <= CUTOFF2);                // cutoff filter

    // Pack A (16x4 f32) per ISA layout:
    //   lanes 0-15 : VGPR0 = K0 (x), VGPR1 = K1 (y)   for M = lane
    //   lanes 16-31: VGPR0 = K2 (z), VGPR1 = K3 (0)   for M = lane-16
    // B = A^T of the same data; the matching 4x16 striping carries identical
    // per-lane values, so the same operand feeds both A and B -> Gram matrix.
    v2f ab;
    ab.x = hihalf ? dz   : dx;
    ab.y = hihalf ? 0.0f : dy;

    v8f acc = {};
    // (neg_a, A, neg_b, B, c_mod, C, reuse_a, reuse_b)
    acc = __builtin_amdgcn_wmma_f32_16x16x4_f32(
        false, ab, false, ab, (short)0, acc, false, false);

    // C/D layout: lane L<16 -> (M = v, N = L); lane L>=16 -> (M = v+8, N = L-16)
    const int   N    = slot;
    const float sqN  = sq;
    const bool  valN = val;
    const long long nodeBase = (long long)n * (DEG * DEG);

    #pragma unroll
    for (int v = 0; v < 8; ++v) {
        const int   M     = v + (hihalf ? 8 : 0);
        const float sqM   = __shfl(sq, M, 32);           // slot-M |R1|^2
        const int   valM  = __shfl((int)val, M, 32);     // slot-M cutoff flag
        const int   nbM   = __shfl(nbr, M, 32);          // slot-M neighbor id

        const bool  m = (M != N) && valN && (valM != 0);
        const float d = acc[v];                          // R1_M . R1_N

        // |a x b|^2 = |a|^2 |b|^2 - (a.b)^2  (clamped for fp round-off)
        float cn2 = sqM * sqN - d * d;
        cn2 = cn2 > 0.0f ? cn2 : 0.0f;
        const float cross_norm = sqrtf(m ? cn2 : 1.0f);  // safe sqrt on padded slots
        const float angle = m ? atan2f(cross_norm, d) : 0.0f;

        // |R1_M - R1_N|^2 = |a|^2 + |b|^2 - 2 a.b
        const float d2 = sqM + sqN - 2.0f * d;
        const float djk = m ? ((d2 > 0.0f) ? sqrtf(d2) : 1.0f) : 0.0f;

        const long long idx = nodeBase + (long long)M * DEG + N;
        id3_i[idx]    = n;
        id3_j[idx]    = nbM;
        id3_k[idx]    = nbr;
        dist_out[idx] = djk;
        ang_out[idx]  = angle;
        mask_out[idx] = m ? (unsigned char)1 : (unsigned char)0;
    }
}

extern "C" void kernel_launch(void* const* d_in, const int* in_sizes, int n_in,
                              void* d_out, int out_size, void* d_ws, size_t ws_size,
                              hipStream_t stream) {
    const float* pos        = (const float*)d_in[0];   // [N,3] f32
    const int*   edge_index = (const int*)d_in[1];     // [2,E] i32

    const int num_nodes = in_sizes[0] / 3;
    const long long E3  = (long long)num_nodes * DEG * DEG;   // triplet slots

    // Tuple outputs concatenated flat in return order:
    //   id3_i[i32], id3_j[i32], id3_k[i32], distances[f32], angles[f32], mask[u8]
    char* base = (char*)d_out;
    int*   id3_i    = (int*)base;
    int*   id3_j    = id3_i + E3;
    int*   id3_k    = id3_j + E3;
    float* dist_out = (float*)(id3_k + E3);
    float* ang_out  = dist_out + E3;
    unsigned char* mask_out = (unsigned char*)(ang_out + E3);

    const int threads = 256;                 // 8 wave32 per block, 1 node per wave
    const int waves_per_block = threads / 32;
    const int blocks = (num_nodes + waves_per_block - 1) / waves_per_block;

    triplet_wmma_kernel<<<blocks, threads, 0, stream>>>(
        pos, edge_index, id3_i, id3_j, id3_k, dist_out, ang_out, mask_out, num_nodes);
}